// SamplerAttnFCN_72103910966129
// MI455X (gfx1250) — compile-verified
//
#include <hip/hip_runtime.h>
#include <hip/hip_bf16.h>
#include <float.h>

typedef __attribute__((ext_vector_type(2))) float v2f;
typedef __attribute__((ext_vector_type(8))) float v8f;

#define NNODE 27
#define NF    28            // per-slice feature dim ('f' in the einsum)
#define HBIG  8192          // number of independent slices
#define ROWSTRIDE (NF * HBIG)   // 229376 floats per W_lin row
#define GSTR  29            // LDS row stride for g (avoid bank conflicts)
#define NBLK  1024
#define WAVES 8

// ---------------------------------------------------------------------------
// Kernel 1: one wave per h-slice.
//   g_h (27x28) = node_emb(27x28) @ W_lin[:, h*28:(h+1)*28]  via f32 WMMA
//   s_i/s_j, masked leaky(0.2), softmax over i, weights for row idx,
//   partial x[f] = sum_j w_j * g_h[j,f]; block-reduced into ws.
// ---------------------------------------------------------------------------
__global__ __launch_bounds__(256) void gat_slice_kernel(
    const float* __restrict__ obs,        // 27
    const float* __restrict__ node_emb,   // 27 x 28
    const int*   __restrict__ adj,        // 27 x 27
    const float* __restrict__ W_lin,      // 28 x 229376
    const float* __restrict__ W_attn,     // 56
    const float* __restrict__ b_attn,     // 1
    float* __restrict__ ws_part)          // NBLK x 28
{
    __shared__ float s_ne[32 * NF];           // node_emb zero-padded to 32 rows
    __shared__ int   s_adj[NNODE * NNODE];    // 729
    __shared__ float s_wl[NF];
    __shared__ float s_wr[NF];
    __shared__ float s_g[WAVES][NF * GSTR];   // g rows 0..26 used
    __shared__ float s_si[WAVES][32];
    __shared__ float s_sj[WAVES][32];
    __shared__ float s_w[WAVES][32];
    __shared__ float s_red[WAVES * NF];

    const int tid  = threadIdx.x;
    const int wave = tid >> 5;
    const int lane = tid & 31;

    for (int t = tid; t < 32 * NF; t += 256)
        s_ne[t] = (t < NNODE * NF) ? node_emb[t] : 0.0f;   // zero-pad rows 27..31
    for (int t = tid; t < NNODE * NNODE; t += 256) s_adj[t] = adj[t];
    if (tid < NF) { s_wl[tid] = W_attn[tid]; s_wr[tid] = W_attn[NF + tid]; }
    __syncthreads();

    // idx = argmax(obs != 0) -> first nonzero index (0 if all zero). Uniform.
    int idx = 0;
    for (int n = NNODE - 1; n >= 0; --n)
        if (obs[n] != 0.0f) idx = n;

    const float bat   = b_attn[0];
    const int   slice = blockIdx.x * WAVES + wave;   // 0..8191

    // ---- A fragments: node_emb as 16x4 f32 tiles (M padded 27->32) --------
    // ISA 16x4 f32 A layout: lanes 0-15 hold K={4kt,4kt+1}, lanes 16-31 K={4kt+2,4kt+3}
    const int lhalf = lane >> 4;   // 0 | 1
    const int lm    = lane & 15;
    v2f afrag[2][7];
#pragma unroll
    for (int mt = 0; mt < 2; ++mt) {
        const int m = mt * 16 + lm;           // rows 27..31 read LDS zeros
#pragma unroll
        for (int kt = 0; kt < 7; ++kt) {
            const int k0 = kt * 4 + lhalf * 2;
            v2f a;
            a.x = s_ne[m * NF + k0];
            a.y = s_ne[m * NF + k0 + 1];
            afrag[mt][kt] = a;
        }
    }

    // Padded N columns (28..31) of the C tiles are never stored back, and in
    // D[m,n] only B[:,n] feeds column n, so those B lanes may hold ANY value.
    // Clamp the column index to 0 -> unconditional, branch-free loads.
    const int ncol1 = (16 + lm < NF) ? (16 + lm) : 0;

    // ---- g tile accumulation: 28 x v_wmma_f32_16x16x4_f32 per slice -------
    v8f c00 = {}, c01 = {}, c10 = {}, c11 = {};
    const float* wbase = W_lin + (size_t)slice * NF;
#pragma unroll
    for (int kt = 0; kt < 7; ++kt) {
        const int    k0 = kt * 4 + lhalf * 2;
        const float* r0 = wbase + (size_t)k0       * ROWSTRIDE;
        const float* r1 = wbase + (size_t)(k0 + 1) * ROWSTRIDE;
        // B 4x16 tiles: same K-per-lane-half layout as A; N striped over lanes.
        // W_lin is streamed exactly once across the grid -> non-temporal.
        v2f b0;                       // N-tile 0: columns 0..15
        b0.x = __builtin_nontemporal_load(r0 + lm);
        b0.y = __builtin_nontemporal_load(r1 + lm);
        v2f b1;                       // N-tile 1: columns 16..27 (clamped pad)
        b1.x = __builtin_nontemporal_load(r0 + ncol1);
        b1.y = __builtin_nontemporal_load(r1 + ncol1);

        c00 = __builtin_amdgcn_wmma_f32_16x16x4_f32(false, afrag[0][kt], false, b0, (short)0, c00, false, false);
        c01 = __builtin_amdgcn_wmma_f32_16x16x4_f32(false, afrag[0][kt], false, b1, (short)0, c01, false, false);
        c10 = __builtin_amdgcn_wmma_f32_16x16x4_f32(false, afrag[1][kt], false, b0, (short)0, c10, false, false);
        c11 = __builtin_amdgcn_wmma_f32_16x16x4_f32(false, afrag[1][kt], false, b1, (short)0, c11, false, false);
    }

    // ---- spill g tiles to LDS (C/D layout: M = 16mt + r + 8*lhalf) --------
    float* g = s_g[wave];
#pragma unroll
    for (int r = 0; r < 8; ++r) {
        const int m0 = r + 8 * lhalf;        // tile mt=0: always < 16 < 27
        const int m1 = 16 + r + 8 * lhalf;   // tile mt=1: guard < 27
        g[m0 * GSTR + lm] = c00[r];
        if (16 + lm < NF)  g[m0 * GSTR + 16 + lm] = c01[r];
        if (m1 < NNODE) {
            g[m1 * GSTR + lm] = c10[r];
            if (16 + lm < NF) g[m1 * GSTR + 16 + lm] = c11[r];
        }
    }
    __syncthreads();

    // ---- s_i, s_j ----------------------------------------------------------
    if (lane < NNODE) {
        float si = 0.0f, sj = 0.0f;
#pragma unroll
        for (int f = 0; f < NF; ++f) {
            const float gv = g[lane * GSTR + f];
            si += gv * s_wl[f];
            sj += gv * s_wr[f];
        }
        s_si[wave][lane] = si;
        s_sj[wave][lane] = sj;
    }
    __syncthreads();

    // ---- per-column masked leaky softmax -> weight of row idx -------------
    if (lane < NNODE) {
        const int   j   = lane;
        const float sjv = s_sj[wave][j];
        float m = -FLT_MAX;
        for (int i = 0; i < NNODE; ++i) {
            if (s_adj[i * NNODE + j]) {
                float e = s_si[wave][i] + sjv + bat;
                e = (e >= 0.0f) ? e : 0.2f * e;
                m = fmaxf(m, e);
            }
        }
        float sum = 0.0f, widx = 0.0f;
        for (int i = 0; i < NNODE; ++i) {
            if (s_adj[i * NNODE + j]) {
                float e = s_si[wave][i] + sjv + bat;
                e = (e >= 0.0f) ? e : 0.2f * e;
                const float t = __expf(e - m);
                sum += t;
                if (i == idx) widx = t;      // stays 0 if adj[idx,j]==0
            }
        }
        s_w[wave][j] = widx / sum;           // diagonal guarantees sum > 0
    } else {
        s_w[wave][lane] = 0.0f;
    }
    __syncthreads();

    // ---- x_partial[f] = sum_j w_j * g[j,f] --------------------------------
    if (lane < NF) {
        float acc = 0.0f;
#pragma unroll
        for (int j = 0; j < NNODE; ++j)
            acc += s_w[wave][j] * g[j * GSTR + lane];
        s_red[wave * NF + lane] = acc;
    }
    __syncthreads();

    if (wave == 0 && lane < NF) {
        float tot = 0.0f;
#pragma unroll
        for (int w = 0; w < WAVES; ++w) tot += s_red[w * NF + lane];
        ws_part[blockIdx.x * NF + lane] = tot;
    }
}

// ---------------------------------------------------------------------------
// Kernel 2: reduce 1024x28 partials, mean over 8192 slices, 28->256->5 MLP.
// ---------------------------------------------------------------------------
__global__ __launch_bounds__(256) void gat_tail_kernel(
    const float* __restrict__ ws_part,    // NBLK x 28
    const float* __restrict__ W1,         // 28 x 256
    const float* __restrict__ b1,         // 256
    const float* __restrict__ W2,         // 256 x 5
    const float* __restrict__ b2,         // 5
    float* __restrict__ out)              // 5
{
    __shared__ float part[4 * NF];
    __shared__ float x[NF];
    __shared__ float h[256];
    const int t = threadIdx.x;

    if (t < 4 * NF) {
        const int f = t % NF, grp = t / NF;    // 4 groups x 256 blocks = 1024
        float a = 0.0f;
        for (int b = grp * 256; b < grp * 256 + 256; ++b)
            a += ws_part[b * NF + f];
        part[t] = a;
    }
    __syncthreads();

    if (t < NF)
        x[t] = (part[t] + part[NF + t] + part[2 * NF + t] + part[3 * NF + t]) *
               (1.0f / (float)HBIG);
    __syncthreads();

    {
        float a = b1[t];
#pragma unroll
        for (int f = 0; f < NF; ++f) a += x[f] * W1[f * 256 + t];
        h[t] = (a >= 0.0f) ? a : 0.01f * a;
    }
    __syncthreads();

    if (t < 5) {
        float a = b2[t];
        for (int k = 0; k < 256; ++k) a += h[k] * W2[k * 5 + t];
        out[t] = a;
    }
}

extern "C" void kernel_launch(void* const* d_in, const int* in_sizes, int n_in,
                              void* d_out, int out_size, void* d_ws, size_t ws_size,
                              hipStream_t stream) {
    const float* obs      = (const float*)d_in[0];
    const float* node_emb = (const float*)d_in[1];
    const int*   adj      = (const int*)  d_in[2];
    const float* W_lin    = (const float*)d_in[3];
    const float* W_attn   = (const float*)d_in[4];
    const float* b_attn   = (const float*)d_in[5];
    const float* W1       = (const float*)d_in[6];
    const float* b1       = (const float*)d_in[7];
    const float* W2       = (const float*)d_in[8];
    const float* b2       = (const float*)d_in[9];
    float* ws = (float*)d_ws;    // needs NBLK*28*4 = 114688 bytes

    gat_slice_kernel<<<NBLK, 256, 0, stream>>>(obs, node_emb, adj, W_lin,
                                               W_attn, b_attn, ws);
    gat_tail_kernel<<<1, 256, 0, stream>>>(ws, W1, b1, W2, b2, (float*)d_out);
}